// DynamicRepresentationAlignment_65154653880841
// MI455X (gfx1250) — compile-verified
//
#include <hip/hip_runtime.h>

typedef float v2f __attribute__((ext_vector_type(2)));
typedef float v8f __attribute__((ext_vector_type(8)));

#define EPSN 1e-12f

constexpr int Bn = 4, Cn = 8, Dn = 256, HWn = 256 * 256;
constexpr int NSEG = 8;              // HW segments per (b, d-tile)
constexpr int NTILED = Dn / 16;      // 16 d-tiles
constexpr int NTILES = Bn * NTILED;  // 64 output tiles
constexpr int WAVES = 8;             // 256-thread block = 8 wave32
constexpr int HW_PER_WAVE = HWn / (NSEG * WAVES); // 1024

// Phase 1: region_part[(tile*NSEG+seg)*256 + pos] = partial 16x16 (d_local x c) tile
//          msum_part[(b*NSEG+seg)*16 + c]         = partial mask sums (dt==0 blocks only)
__global__ __launch_bounds__(256) void dra_phase1(const float* __restrict__ feat,
                                                  const float* __restrict__ mask,
                                                  float* __restrict__ region_part,
                                                  float* __restrict__ msum_part) {
  const int blk  = blockIdx.x;
  const int seg  = blk % NSEG;
  const int dt   = (blk / NSEG) % NTILED;
  const int b    = blk / (NSEG * NTILED);
  const int tid  = threadIdx.x;
  const int wave = tid >> 5;
  const int lane = tid & 31;
  const int lane_lo = lane & 15;   // = M row (d_local) for A, = N col (c) for B
  const int lane_hi = lane >> 4;   // selects K-pair {0,1} vs {2,3}

  const int hw0 = (seg * WAVES + wave) * HW_PER_WAVE;

  // A (features): lane reads 2 contiguous floats of row d = dt*16 + lane_lo
  const float* fp = feat + (size_t)(b * Dn + dt * 16 + lane_lo) * HWn + hw0 + 2 * lane_hi;
  // B (masks): lane reads 2 contiguous floats of row c = lane_lo (<8), else zeroed
  const int c = lane_lo;
  const float cs = (c < Cn) ? 1.0f : 0.0f;
  const float* mp = mask + (size_t)(b * Cn + (c < Cn ? c : 0)) * HWn + hw0 + 2 * lane_hi;

  v8f acc = {0.f, 0.f, 0.f, 0.f, 0.f, 0.f, 0.f, 0.f};
  float bsum = 0.f;

#pragma unroll 16
  for (int k = 0; k < HW_PER_WAVE; k += 4) {
    v2f a  = *(const v2f*)(fp + k);
    v2f bm = *(const v2f*)(mp + k);
    bm.x *= cs;                 // zero lanes for c >= 8 (no divergence; EXEC stays full)
    bm.y *= cs;
    bsum += bm.x + bm.y;        // free msum partial from data already in registers
    // D(16x16, d_local x c) += A(16x4 f32) * B(4x16 f32), full fp32 precision
    acc = __builtin_amdgcn_wmma_f32_16x16x4_f32(false, a, false, bm,
                                                (short)0, acc, false, false);
  }

  // Reduce 8 waves -> one tile via LDS float atomics (ds_add_f32)
  __shared__ float ltile[256];   // [v*32 + lane]: element (d_local = v + 8*(lane/16), c = lane%16)
  __shared__ float lmsum[16];
  ltile[tid] = 0.f;
  if (tid < 16) lmsum[tid] = 0.f;
  __syncthreads();
#pragma unroll
  for (int v = 0; v < 8; ++v) atomicAdd(&ltile[v * 32 + lane], acc[v]);
  atomicAdd(&lmsum[lane_lo], bsum);
  __syncthreads();

  region_part[((size_t)(b * NTILED + dt) * NSEG + seg) * 256 + tid] = ltile[tid];
  if (dt == 0 && tid < 16) msum_part[(b * NSEG + seg) * 16 + tid] = lmsum[tid];
}

// Phase 2a: parallel collapse of segment partials -> compact region tiles (64 blocks)
__global__ __launch_bounds__(256) void dra_phase2a(const float* __restrict__ region_part,
                                                   float* __restrict__ region) {
  const int tile = blockIdx.x;       // 0..63  (= b*16 + dt)
  const int tid  = threadIdx.x;
  float r = 0.f;
#pragma unroll
  for (int s = 0; s < NSEG; ++s)
    r += region_part[((size_t)tile * NSEG + s) * 256 + tid];
  region[(size_t)tile * 256 + tid] = r;
}

// Phase 2b: single block; pair = (b,c) handled by 8 threads, each covering 32 d-values.
__global__ __launch_bounds__(256) void dra_phase2b(const float* __restrict__ region,
                                                   const float* __restrict__ msum_part,
                                                   const float* __restrict__ memv,
                                                   float* __restrict__ out) {
  const int tid  = threadIdx.x;
  const int pair = tid >> 3;   // 0..31
  const int sub  = tid & 7;
  const int b    = pair >> 3;  // 0..3
  const int c    = pair & 7;

  float msum = 0.f;
#pragma unroll
  for (int s = 0; s < NSEG; ++s) msum += msum_part[(b * NSEG + s) * 16 + c];
  const float inv = 1.0f / fmaxf(msum, 1.0f);

  float rr = 0.f, rm = 0.f, mm = 0.f;
  for (int i = 0; i < 32; ++i) {
    const int d  = sub * 32 + i;
    const int dt = d >> 4, dl = d & 15;
    const int pos = (dl & 7) * 32 + (dl >> 3) * 16 + c;  // WMMA C/D layout
    float r = region[(size_t)(b * NTILED + dt) * 256 + pos] * inv;
    const float m = memv[c * Dn + d];
    rr += r * r; rm += r * m; mm += m * m;
  }

  __shared__ float srr[256], srm[256], smm[256];
  __shared__ float scon[32];
  __shared__ int   svld[32];
  srr[tid] = rr; srm[tid] = rm; smm[tid] = mm;
  __syncthreads();
  if (sub == 0) {
    for (int i = 1; i < 8; ++i) { rr += srr[tid + i]; rm += srm[tid + i]; mm += smm[tid + i]; }
    const float cosv = rm / (fmaxf(sqrtf(rr), EPSN) * fmaxf(sqrtf(mm), EPSN));
    const bool valid = msum > 0.f;
    scon[pair] = valid ? (1.0f - cosv) : 0.f;
    svld[pair] = valid ? 1 : 0;
  }
  __syncthreads();
  if (tid == 0) {
    float total = 0.f; int cnt = 0;
    for (int p = 0; p < 32; ++p) { total += scon[p]; cnt += svld[p]; }
    out[0] = total / (float)(cnt > 1 ? cnt : 1);
  }
}

extern "C" void kernel_launch(void* const* d_in, const int* in_sizes, int n_in,
                              void* d_out, int out_size, void* d_ws, size_t ws_size,
                              hipStream_t stream) {
  const float* feat = (const float*)d_in[0];   // (4, 256, 256, 256)
  const float* mask = (const float*)d_in[1];   // (4, 8, 256, 256)
  const float* memv = (const float*)d_in[2];   // (8, 256)

  float* region_part = (float*)d_ws;                                   // 131072 floats
  float* msum_part   = region_part + (size_t)NTILES * NSEG * 256;      // 512 floats
  float* region      = msum_part + (size_t)Bn * NSEG * 16;             // 16384 floats

  dra_phase1<<<dim3(Bn * NTILED * NSEG), dim3(256), 0, stream>>>(feat, mask,
                                                                 region_part, msum_part);
  dra_phase2a<<<dim3(NTILES), dim3(256), 0, stream>>>(region_part, region);
  dra_phase2b<<<dim3(1), dim3(256), 0, stream>>>(region, msum_part, memv,
                                                 (float*)d_out);
}